// EdgeUpdateNetwork_46677704573339
// MI455X (gfx1250) — compile-verified
//
#include <hip/hip_runtime.h>

typedef __attribute__((ext_vector_type(2))) float v2f;
typedef __attribute__((ext_vector_type(4))) float v4f;
typedef __attribute__((ext_vector_type(8))) float v8f;

#define T_DIM 8
#define N_DIM 256
#define F_DIM 128
#define TN    (T_DIM * N_DIM)      // 2048 rows
#define BN_EPS 1e-5f
#define L1_EPS 1e-12f

// ---------------------------------------------------------------------------
// Kernel 1: P[mat][m,f] = X[m,:] . Wpart[f,:]   via V_WMMA_F32_16X16X4_F32
//   mat 0 = w1[:,0:128], 1 = w1[:,128:256], 2 = w2[:,0:128], 3 = w2[:,128:256]
// One wave per 16x16 D tile; K = 128 consumed 4 at a time (32 WMMA ops).
// Lane layout per CDNA5 ISA 7.12.2:
//   A 16x4 : lanes 0-15 -> M=lane, V0=K0,V1=K1 ; lanes 16-31 -> V0=K2,V1=K3
//   B 4x16 : mirrored (lane&15 = N, lane>=16 selects K pair 2/3)
//   C/D    : lane&15 = N, VGPR v holds M = v + 8*(lane>=16)
// ---------------------------------------------------------------------------
__global__ __launch_bounds__(32)
void gemm_wmma(const float* __restrict__ X,
               const float* __restrict__ w1,
               const float* __restrict__ w2,
               float* __restrict__ P)
{
    const int lane  = threadIdx.x;
    const int nBase = blockIdx.x * 16;           // output feature tile (f)
    const int mBase = blockIdx.y * 16;           // row tile (t*N + i)
    const int mat   = blockIdx.z;                // which weight half

    const float* w      = (mat < 2) ? w1 : w2;
    const int    colOff = (mat & 1) * F_DIM;
    float*       Pm     = P + (size_t)mat * TN * F_DIM;

    const int lm = lane & 15;
    const int kg = (lane >> 4) << 1;             // 0 or 2

    const float* arow = X + (size_t)(mBase + lm) * F_DIM;
    const float* brow = w + (size_t)(nBase + lm) * (2 * F_DIM) + colOff;

    v8f c = {0.f, 0.f, 0.f, 0.f, 0.f, 0.f, 0.f, 0.f};
    for (int k0 = 0; k0 < F_DIM; k0 += 4) {
        v2f a, b;
        a.x = arow[k0 + kg];
        a.y = arow[k0 + kg + 1];
        b.x = brow[k0 + kg];
        b.y = brow[k0 + kg + 1];
        // 8 args: (neg_a, A, neg_b, B, c_mod, C, reuse_a, reuse_b)
        c = __builtin_amdgcn_wmma_f32_16x16x4_f32(
                false, a, false, b, (short)0, c, false, false);
    }

    const int n    = nBase + lm;
    const int mOff = (lane >> 4) << 3;           // 0 or 8
#pragma unroll
    for (int v = 0; v < 8; ++v)
        Pm[(size_t)(mBase + mOff + v) * F_DIM + n] = c[v];
}

// ---------------------------------------------------------------------------
// Kernel 2: exact BN stats in factored form.
//   mean  = (sumA + sumB) / (T*N)
//   E[y2] = sum_t( N*(sumA2_t + sumB2_t) + 2*sumA_t*sumB_t ) / (T*N*N)
// One block of 256 threads per (branch, f); per-t LDS tree reductions.
// Emits fused scale/shift: z = y*scale + shift.
// ---------------------------------------------------------------------------
__global__ __launch_bounds__(256)
void bn_stats(const float* __restrict__ P,
              const float* __restrict__ g1, const float* __restrict__ b1,
              const float* __restrict__ g2, const float* __restrict__ b2,
              float* __restrict__ scaleArr, float* __restrict__ shiftArr)
{
    const int br  = blockIdx.x >> 7;             // branch 0/1
    const int f   = blockIdx.x & 127;
    const int tid = threadIdx.x;                 // == i in [0,256)

    const float* A = P + (size_t)(2 * br)     * TN * F_DIM;
    const float* B = P + (size_t)(2 * br + 1) * TN * F_DIM;

    __shared__ float sa[256], sa2[256], sb[256], sb2[256];
    __shared__ float acc[5];                     // totA, totA2, totB, totB2, cross
    if (tid < 5) acc[tid] = 0.f;
    __syncthreads();

    for (int t = 0; t < T_DIM; ++t) {
        float a = A[(size_t)(t * N_DIM + tid) * F_DIM + f];
        float b = B[(size_t)(t * N_DIM + tid) * F_DIM + f];
        sa[tid] = a;  sa2[tid] = a * a;
        sb[tid] = b;  sb2[tid] = b * b;
        __syncthreads();
        for (int s = 128; s > 0; s >>= 1) {
            if (tid < s) {
                sa[tid]  += sa[tid + s];  sa2[tid] += sa2[tid + s];
                sb[tid]  += sb[tid + s];  sb2[tid] += sb2[tid + s];
            }
            __syncthreads();
        }
        if (tid == 0) {
            acc[0] += sa[0];  acc[1] += sa2[0];
            acc[2] += sb[0];  acc[3] += sb2[0];
            acc[4] += sa[0] * sb[0];             // per-t cross term
        }
        __syncthreads();
    }

    if (tid == 0) {
        const float invTN  = 1.f / (float)(T_DIM * N_DIM);
        const float invTNN = 1.f / ((float)T_DIM * (float)N_DIM * (float)N_DIM);
        float mean = (acc[0] + acc[2]) * invTN;
        float e2   = ((float)N_DIM * (acc[1] + acc[3]) + 2.f * acc[4]) * invTNN;
        float var  = e2 - mean * mean;
        float g    = (br == 0) ? g1[f] : g2[f];
        float bb   = (br == 0) ? b1[f] : b2[f];
        float scale = g * rsqrtf(var + BN_EPS);
        scaleArr[br * F_DIM + f] = scale;
        shiftArr[br * F_DIM + f] = bb - mean * scale;
    }
}

// ---------------------------------------------------------------------------
// Kernel 3: streaming output. One wave32 per output row (t,br,i,j): each lane
// handles 4 consecutive f (16B ld/st), fused affine+ReLU, L1 sum via
// __shfl_xor wave reduction, normalize, store.
// Output is 512 MiB (write-once, never re-read on device, > 192 MB L2) ->
// store NON-TEMPORAL so the reused P matrices (4 MiB) stay resident in L2.
// Floor: ~22us at 23.3 TB/s HBM write bandwidth.
// ---------------------------------------------------------------------------
__global__ __launch_bounds__(256)
void edge_out(const float* __restrict__ P,
              const float* __restrict__ scaleArr,
              const float* __restrict__ shiftArr,
              float* __restrict__ out)
{
    const int lane = threadIdx.x & 31;
    const int wave = threadIdx.x >> 5;
    const long long row = (long long)blockIdx.x * 8 + wave;  // [0, T*2*N*N)

    const int j  = (int)(row & 255);
    const int i  = (int)((row >> 8) & 255);
    const int br = (int)((row >> 16) & 1);
    const int t  = (int)(row >> 17);
    const int f  = lane * 4;

    v4f* dst = (v4f*)(out + row * F_DIM + f);

    if (i == j) {                                 // uniform across the wave
        v4f zero = {0.f, 0.f, 0.f, 0.f};
        __builtin_nontemporal_store(zero, dst);
        return;
    }

    const v4f a  = *(const v4f*)(P + (size_t)(2 * br)     * TN * F_DIM
                                   + (size_t)(t * N_DIM + i) * F_DIM + f);
    const v4f b  = *(const v4f*)(P + (size_t)(2 * br + 1) * TN * F_DIM
                                   + (size_t)(t * N_DIM + j) * F_DIM + f);
    const v4f sc = *(const v4f*)(scaleArr + br * F_DIM + f);
    const v4f sh = *(const v4f*)(shiftArr + br * F_DIM + f);

    float z0 = fmaxf((a.x + b.x) * sc.x + sh.x, 0.f);
    float z1 = fmaxf((a.y + b.y) * sc.y + sh.y, 0.f);
    float z2 = fmaxf((a.z + b.z) * sc.z + sh.z, 0.f);
    float z3 = fmaxf((a.w + b.w) * sc.w + sh.w, 0.f);

    float s = z0 + z1 + z2 + z3;                  // |z| == z after ReLU
#pragma unroll
    for (int m = 16; m >= 1; m >>= 1)
        s += __shfl_xor(s, m, 32);

    float inv = 1.f / fmaxf(s, L1_EPS);
    v4f res = {z0 * inv, z1 * inv, z2 * inv, z3 * inv};
    __builtin_nontemporal_store(res, dst);
}

// ---------------------------------------------------------------------------
extern "C" void kernel_launch(void* const* d_in, const int* in_sizes, int n_in,
                              void* d_out, int out_size, void* d_ws, size_t ws_size,
                              hipStream_t stream)
{
    const float* X  = (const float*)d_in[0];   // [T,N,F]
    const float* w1 = (const float*)d_in[1];   // [F,2F]
    const float* g1 = (const float*)d_in[2];
    const float* b1 = (const float*)d_in[3];
    const float* w2 = (const float*)d_in[4];
    const float* g2 = (const float*)d_in[5];
    const float* b2 = (const float*)d_in[6];
    float* out = (float*)d_out;                // [T,2,N,N,F]

    // Workspace: P = 4 * 2048*128 floats (4 MiB), then scale/shift (2*128 each).
    float* P        = (float*)d_ws;
    float* scaleArr = P + (size_t)4 * TN * F_DIM;
    float* shiftArr = scaleArr + 2 * F_DIM;

    dim3 gGemm(F_DIM / 16, TN / 16, 4);        // 8 x 128 x 4 tiles
    gemm_wmma<<<gGemm, 32, 0, stream>>>(X, w1, w2, P);

    bn_stats<<<2 * F_DIM, 256, 0, stream>>>(P, g1, b1, g2, b2, scaleArr, shiftArr);

    const int totalRows = T_DIM * 2 * N_DIM * N_DIM;   // 1,048,576
    edge_out<<<totalRows / 8, 256, 0, stream>>>(P, scaleArr, shiftArr, out);
}